// CartTensorOut_89764816486666
// MI455X (gfx1250) — compile-verified
//
#include <hip/hip_runtime.h>

// ---------------------------------------------------------------------------
// CDNA5 / gfx1250 fused equivariant-GNN readout kernel.
// All dense GEMMs use V_WMMA_F32_16X16X4_F32 (fp32 matches reference).
// One wave32 per block handles a 16-node tile end-to-end through LDS.
// Round-2 changes: (a) dropped per-fragment tail masking (invalid A rows only
// affect invalid D rows, which are never consumed; loads stay in-bounds via a
// clamped row index), (b) dual WMMA accumulators per GEMM to break the serial
// RAW chain on C and keep the matrix pipe busy.
// ---------------------------------------------------------------------------

typedef __attribute__((ext_vector_type(2))) float v2f;
typedef __attribute__((ext_vector_type(8))) float v8f;

#define TPB 32  // one wave per block

#define INV_SQRT128 0.088388347648318447f
#define INV_SQRT96  0.102062072615965754f
#define INV_SQRT64  0.125f
#define INV_SQRT32  0.176776695296636893f
#define INV_SQRT3   0.577350269189625765f
#define INV_SQRT5   0.447213595499957939f
// Real-CG normalizations (isometric-map construction; CG112/CG011/CG022/
// CG(l,l,0) are exact; these three are the remaining overall scales).
#define C121  0.774596669241483377f   /* sqrt(3/5): l=1 part of 1x2 */
#define C122  0.707106781186547524f   /* l=2 part of 1x2 */
#define C222 -0.534522483824848770f   /* 2x2 -> 2 */

#define RED(x) ((x) + __shfl_xor((x), 16, 32))

__device__ __forceinline__ v8f z8() {
  v8f v;
#pragma unroll
  for (int i = 0; i < 8; ++i) v[i] = 0.f;
  return v;
}

__device__ __forceinline__ v8f wmma4(v2f a, v2f b, v8f c) {
  // D = A(16x4,f32) * B(4x16,f32) + C(16x16,f32)
  return __builtin_amdgcn_wmma_f32_16x16x4_f32(false, a, false, b, (short)0, c,
                                               false, false);
}

// ----- small symmetric 3x3 helpers; order (xx,yy,zz,xy,yz,xz) --------------
struct Uv { float u0, vx, vy, vz, s[5]; };

__device__ __forceinline__ Uv loadU(const float* uSw, int nd, int c) {
  const float* p = uSw + (nd * 32 + c) * 9;
  Uv u;
  u.u0 = p[0];
  u.vy = p[1]; u.vz = p[2]; u.vx = p[3];  // x1 component order is (y,z,x)
#pragma unroll
  for (int j = 0; j < 5; ++j) u.s[j] = p[4 + j];
  return u;
}

__device__ __forceinline__ float dot5(const float* s) {
  return s[0]*s[0] + s[1]*s[1] + s[2]*s[2] + s[3]*s[3] + s[4]*s[4];
}
// isometric 5-vec (xy,yz,z2,xz,x2-y2) -> sym traceless 3x3
__device__ __forceinline__ void mat3from5(const float* s, float* M) {
  const float i6 = 0.408248290463863016f, i2 = 0.707106781186547524f;
  M[0] = -s[2]*i6 + s[4]*i2;
  M[1] = -s[2]*i6 - s[4]*i2;
  M[2] =  2.f*s[2]*i6;
  M[3] =  s[0]*i2;  M[4] = s[1]*i2;  M[5] = s[3]*i2;
}
// inverse map (projects out trace automatically)
__device__ __forceinline__ void vec5fromSym(const float* M, float* s) {
  const float s2 = 1.414213562373095049f, i6 = 0.408248290463863016f;
  s[0] = s2*M[3];
  s[1] = s2*M[4];
  s[2] = (2.f*M[2] - M[0] - M[1]) * i6;
  s[3] = s2*M[5];
  s[4] = (M[0] - M[1]) * (1.f/s2);
}
__device__ __forceinline__ void symSq(const float* M, float* R) {
  R[0] = M[0]*M[0] + M[3]*M[3] + M[5]*M[5];
  R[1] = M[3]*M[3] + M[1]*M[1] + M[4]*M[4];
  R[2] = M[5]*M[5] + M[4]*M[4] + M[2]*M[2];
  R[3] = M[0]*M[3] + M[3]*M[1] + M[5]*M[4];
  R[4] = M[3]*M[5] + M[1]*M[4] + M[4]*M[2];
  R[5] = M[0]*M[5] + M[3]*M[4] + M[5]*M[2];
}
__device__ __forceinline__ void symMv(const float* M, float vx, float vy,
                                      float vz, float* w) {
  w[0] = M[0]*vx + M[3]*vy + M[5]*vz;
  w[1] = M[3]*vx + M[1]*vy + M[4]*vz;
  w[2] = M[5]*vx + M[4]*vy + M[2]*vz;
}
// l=2 part of vector x sym-traceless: P = sym(v x columns(M))  (traceless)
__device__ __forceinline__ void crossP(float vx, float vy, float vz,
                                       const float* M, float* P) {
  float A00 = vy*M[5] - vz*M[3];
  float A10 = vz*M[0] - vx*M[5];
  float A20 = vx*M[3] - vy*M[0];
  float A01 = vy*M[4] - vz*M[1];
  float A11 = vz*M[3] - vx*M[4];
  float A21 = vx*M[1] - vy*M[3];
  float A02 = vy*M[2] - vz*M[4];
  float A12 = vz*M[5] - vx*M[2];
  float A22 = vx*M[4] - vy*M[5];
  P[0] = A00; P[1] = A11; P[2] = A22;
  P[3] = 0.5f*(A01 + A10);
  P[4] = 0.5f*(A12 + A21);
  P[5] = 0.5f*(A02 + A20);
}

// ---------------------------------------------------------------------------
__global__ __launch_bounds__(TPB) void cart_tensor_kernel(
    const float* __restrict__ nsc, const float* __restrict__ equi,
    const int* __restrict__ batch,
    const float* __restrict__ W0, const float* __restrict__ W1p,
    const float* __restrict__ W2p,
    const float* __restrict__ g0e, const float* __restrict__ g1o,
    const float* __restrict__ g2e, const float* __restrict__ g2o,
    const float* __restrict__ mw1, const float* __restrict__ mb1,
    const float* __restrict__ mw2, const float* __restrict__ mb2,
    const float* __restrict__ p0, const float* __restrict__ p2,
    float* __restrict__ out, int N) {
  __shared__ float uS[16 * 32 * 9];   // 18 KB: u0|u1(yzx)|u2(5) per node/ch
  __shared__ float hS[16 * 64];       //  4 KB: silu MLP hidden
  __shared__ float twS[16 * 576];     // 36 KB: packed live tw columns

  const int lane = threadIdx.x & 31;
  const int mrow = lane & 15;         // WMMA A row / node-in-tile
  const int half = lane >> 4;         // K-half for A/B fragments
  const int kh   = half * 2;
  const int ncol = lane & 15;         // WMMA B/D column

  const int n0 = blockIdx.x * 16;
  const int nr = n0 + mrow;
  // Clamped row keeps loads in-bounds; invalid rows produce rows of D that
  // are never consumed (atomics are node-guarded), so no masking needed.
  const size_t nrc = (size_t)((nr < N) ? nr : 0);

  float* uSw = uS; float* hSw = hS; float* twSw = twS;

  // ================= Phase A: WMMA GEMMs -> LDS =================
  // u0 = x0 @ W0 / sqrt(128)           (K=128, 32 out channels)
  for (int nt = 0; nt < 2; ++nt) {
    v8f ac0 = z8(), ac1 = z8();
#pragma unroll 4
    for (int k0 = 0; k0 < 128; k0 += 8) {
      v2f a, b;
      a.x = equi[nrc*475 + (k0+kh)];
      a.y = equi[nrc*475 + (k0+kh+1)];
      b.x = W0[(k0+kh)*32   + nt*16 + ncol];
      b.y = W0[(k0+kh+1)*32 + nt*16 + ncol];
      ac0 = wmma4(a, b, ac0);
      a.x = equi[nrc*475 + (k0+4+kh)];
      a.y = equi[nrc*475 + (k0+4+kh+1)];
      b.x = W0[(k0+4+kh)*32   + nt*16 + ncol];
      b.y = W0[(k0+4+kh+1)*32 + nt*16 + ncol];
      ac1 = wmma4(a, b, ac1);
    }
#pragma unroll
    for (int r = 0; r < 8; ++r)
      uSw[((r + 8*half)*32 + nt*16 + ncol)*9 + 0] =
          (ac0[r] + ac1[r]) * INV_SQRT128;
  }
  // u1_i = x1[:,:,i] @ W1p / sqrt(64)  (K=64, strided A, 3 components)
  for (int i = 0; i < 3; ++i)
    for (int nt = 0; nt < 2; ++nt) {
      v8f ac0 = z8(), ac1 = z8();
#pragma unroll 4
      for (int k0 = 0; k0 < 64; k0 += 8) {
        v2f a, b;
        a.x = equi[nrc*475 + 128 + (k0+kh)*3   + i];
        a.y = equi[nrc*475 + 128 + (k0+kh+1)*3 + i];
        b.x = W1p[(k0+kh)*32   + nt*16 + ncol];
        b.y = W1p[(k0+kh+1)*32 + nt*16 + ncol];
        ac0 = wmma4(a, b, ac0);
        a.x = equi[nrc*475 + 128 + (k0+4+kh)*3   + i];
        a.y = equi[nrc*475 + 128 + (k0+4+kh+1)*3 + i];
        b.x = W1p[(k0+4+kh)*32   + nt*16 + ncol];
        b.y = W1p[(k0+4+kh+1)*32 + nt*16 + ncol];
        ac1 = wmma4(a, b, ac1);
      }
#pragma unroll
      for (int r = 0; r < 8; ++r)
        uSw[((r + 8*half)*32 + nt*16 + ncol)*9 + 1 + i] =
            (ac0[r] + ac1[r]) * INV_SQRT64;
    }
  // u2_i = x2[:,:,i] @ W2p / sqrt(32)  (K=32, 5 comps; guard 475-col row end)
  for (int i = 0; i < 5; ++i)
    for (int nt = 0; nt < 2; ++nt) {
      v8f ac0 = z8(), ac1 = z8();
#pragma unroll
      for (int k0 = 0; k0 < 32; k0 += 8) {
        int c0 = 320 + (k0+kh)*5 + i, c1 = 320 + (k0+kh+1)*5 + i;
        v2f a, b;
        a.x = (c0 < 475) ? equi[nrc*475 + c0] : 0.f;
        a.y = (c1 < 475) ? equi[nrc*475 + c1] : 0.f;
        b.x = W2p[(k0+kh)*32   + nt*16 + ncol];
        b.y = W2p[(k0+kh+1)*32 + nt*16 + ncol];
        ac0 = wmma4(a, b, ac0);
        c0 = 320 + (k0+4+kh)*5 + i; c1 = 320 + (k0+4+kh+1)*5 + i;
        a.x = (c0 < 475) ? equi[nrc*475 + c0] : 0.f;
        a.y = (c1 < 475) ? equi[nrc*475 + c1] : 0.f;
        b.x = W2p[(k0+4+kh)*32   + nt*16 + ncol];
        b.y = W2p[(k0+4+kh+1)*32 + nt*16 + ncol];
        ac1 = wmma4(a, b, ac1);
      }
#pragma unroll
      for (int r = 0; r < 8; ++r)
        uSw[((r + 8*half)*32 + nt*16 + ncol)*9 + 4 + i] =
            (ac0[r] + ac1[r]) * INV_SQRT32;
    }
  // h = silu(node_scalar @ mw1 / sqrt(128) + mb1)   (K=128, 64 channels)
  for (int nt = 0; nt < 4; ++nt) {
    v8f ac0 = z8(), ac1 = z8();
#pragma unroll 4
    for (int k0 = 0; k0 < 128; k0 += 8) {
      v2f a, b;
      a.x = nsc[nrc*128 + k0+kh];
      a.y = nsc[nrc*128 + k0+kh+1];
      b.x = mw1[(k0+kh)*64   + nt*16 + ncol];
      b.y = mw1[(k0+kh+1)*64 + nt*16 + ncol];
      ac0 = wmma4(a, b, ac0);
      a.x = nsc[nrc*128 + k0+4+kh];
      a.y = nsc[nrc*128 + k0+4+kh+1];
      b.x = mw1[(k0+4+kh)*64   + nt*16 + ncol];
      b.y = mw1[(k0+4+kh+1)*64 + nt*16 + ncol];
      ac1 = wmma4(a, b, ac1);
    }
    const float bb = mb1[nt*16 + ncol];
#pragma unroll
    for (int r = 0; r < 8; ++r) {
      const float x = (ac0[r] + ac1[r])*INV_SQRT128 + bb;
      hSw[(r + 8*half)*64 + nt*16 + ncol] = x / (1.f + __expf(-x));
    }
  }
  __syncthreads();

  // ====== Phase B: stage the 576 live tw columns (of 960) via WMMA ======
  // Entries with antisymmetric CG (111/221) on proportional operands are
  // identically zero and skipped; offsets keep the reference tw layout.
  {
    const int srcB[8] = {0, 96, 224, 320, 608, 736, 832, 864};
    const int dstB[8] = {0, 96, 160, 256, 288, 352, 448, 480};
    const int ntl[8]  = {6, 4, 6, 2, 4, 6, 2, 6};
#pragma unroll
    for (int e = 0; e < 8; ++e)
      for (int t = 0; t < ntl[e]; ++t) {
        const int sb = srcB[e] + t*16, db = dstB[e] + t*16;
        v8f ac0 = z8(), ac1 = z8();
#pragma unroll 4
        for (int k0 = 0; k0 < 64; k0 += 8) {
          v2f a, b;
          a.x = hSw[mrow*64 + k0+kh];
          a.y = hSw[mrow*64 + k0+kh+1];
          b.x = mw2[(size_t)(k0+kh)*960   + sb + ncol];
          b.y = mw2[(size_t)(k0+kh+1)*960 + sb + ncol];
          ac0 = wmma4(a, b, ac0);
          a.x = hSw[mrow*64 + k0+4+kh];
          a.y = hSw[mrow*64 + k0+4+kh+1];
          b.x = mw2[(size_t)(k0+4+kh)*960   + sb + ncol];
          b.y = mw2[(size_t)(k0+4+kh+1)*960 + sb + ncol];
          ac1 = wmma4(a, b, ac1);
        }
        const float bb = mb2[sb + ncol];
#pragma unroll
        for (int r = 0; r < 8; ++r)
          twSw[(r + 8*half)*576 + db + ncol] =
              (ac0[r] + ac1[r])*INV_SQRT64 + bb;
      }
  }
  __syncthreads();

  // ====== Phase C: equivariant algebra; lane pair (l, l+16) = node l&15 ===
  // Pass 1: eq_ln norms (g excluded by definition)
  float acc0 = 0.f, acc1 = 0.f, acc2 = 0.f, acc3 = 0.f;
  for (int t = 0; t < 16; ++t) {
    const int c = 2*t + half;
    Uv u = loadU(uSw, mrow, c);
    const float d11 = u.vx*u.vx + u.vy*u.vy + u.vz*u.vz;
    const float d22 = dot5(u.s);
    float M[6]; mat3from5(u.s, M);
    float R[6]; symSq(M, R);
    float q5[5]; vec5fromSym(R, q5);
    float wv3[3]; symMv(M, u.vx, u.vy, u.vz, wv3);
    const float mv2 = wv3[0]*wv3[0] + wv3[1]*wv3[1] + wv3[2]*wv3[2];
    float P[6]; crossP(u.vx, u.vy, u.vz, M, P);
    float p5v[5]; vec5fromSym(P, p5v);
    const float m00 = u.u0*u.u0, m01 = d11*INV_SQRT3, m02 = d22*INV_SQRT5;
    acc0 += m00*m00 + m01*m01 + m02*m02;                       // m0e
    acc1 += m00*d11 + C121*C121*mv2;                           // m1o
    acc2 += m00*d22 + (2.f/3.f)*d11*d11 + C222*C222*dot5(q5);  // m2e
    acc3 += C122*C122*dot5(p5v);                               // m2o
  }
  const float r0  = rsqrtf(RED(acc0)/96.f + 1e-5f);
  const float r1o = rsqrtf(RED(acc1)/64.f + 1e-5f);
  const float r2e = rsqrtf(RED(acc2)/96.f + 1e-5f);
  const float r2o = rsqrtf(RED(acc3)/32.f + 1e-5f);

  // Pass 2: the 8 live weighted pair entries, fused over channels
  float a0acc = 0.f, a1acc = 0.f, a3acc = 0.f, a4acc = 0.f;
  float S9[6]  = {0,0,0,0,0,0};
  float S11[6] = {0,0,0,0,0,0};
  float S12[6] = {0,0,0,0,0,0};
  float A13[5] = {0,0,0,0,0};
  const float* tw = &twSw[mrow*576];
  for (int t = 0; t < 16; ++t) {
    const int c = 2*t + half;
    Uv u = loadU(uSw, mrow, c);
    const float d11 = u.vx*u.vx + u.vy*u.vy + u.vz*u.vz;
    const float d22 = dot5(u.s);
    float M[6]; mat3from5(u.s, M);
    float R[6]; symSq(M, R);
    const float trR3 = (R[0]+R[1]+R[2]) * (1.f/3.f);
    float q5[5]; vec5fromSym(R, q5);
    float wv3[3]; symMv(M, u.vx, u.vy, u.vz, wv3);
    const float mv2 = wv3[0]*wv3[0] + wv3[1]*wv3[1] + wv3[2]*wv3[2];
    float P[6]; crossP(u.vx, u.vy, u.vz, M, P);
    float p5v[5]; vec5fromSym(P, p5v);
    const float O[6] = {u.vx*u.vx, u.vy*u.vy, u.vz*u.vz,
                        u.vx*u.vy, u.vy*u.vz, u.vx*u.vz};
    float qv[5]; vec5fromSym(O, qv);

    const float f0a = r0*g0e[c],  f0b = r0*g0e[32+c],  f0c = r0*g0e[64+c];
    const float f1a = r1o*g1o[c], f1b = r1o*g1o[32+c];
    const float f2a = r2e*g2e[c], f2b = r2e*g2e[32+c], f2c = r2e*g2e[64+c];
    const float f2o = r2o*g2o[c];
    const float m0a = u.u0*u.u0*f0a, m0b = d11*INV_SQRT3*f0b,
                m0c = d22*INV_SQRT5*f0c;

    // (m0e,m0e,000)
    a0acc += tw[c]*m0a*m0a + tw[32+c]*m0b*m0b + tw[64+c]*m0c*m0c;
    // (m1o,m1o,110)
    a1acc += tw[96+c]    * f1a*f1a * u.u0*u.u0 * d11;
    a1acc += tw[96+32+c] * f1b*f1b * C121*C121 * mv2;
    // (m2e,m2e,220)
    a3acc += tw[160+c]    * f2a*f2a * u.u0*u.u0 * d22;
    a3acc += tw[160+32+c] * f2b*f2b * (2.f/3.f) * d11*d11;
    a3acc += tw[160+64+c] * f2c*f2c * C222*C222 * dot5(q5);
    // (m2o,m2o,220)
    a4acc += tw[256+c] * f2o*f2o * C122*C122 * dot5(p5v);
    // (m1o,m1o,112): outer products of m1o vectors
    {
      const float k1 = tw[288+c]    * f1a*f1a * u.u0*u.u0;
      const float k2 = tw[288+32+c] * f1b*f1b * C121*C121;
#pragma unroll
      for (int j = 0; j < 6; ++j) S9[j] += k1*O[j];
      S9[0]+=k2*wv3[0]*wv3[0]; S9[1]+=k2*wv3[1]*wv3[1]; S9[2]+=k2*wv3[2]*wv3[2];
      S9[3]+=k2*wv3[0]*wv3[1]; S9[4]+=k2*wv3[1]*wv3[2]; S9[5]+=k2*wv3[0]*wv3[2];
    }
    // (m2e,m2e,222): S += w * mat(m)^2 per sub-block
    {
      const float k1 = tw[352+c] * f2a*f2a * u.u0*u.u0;
#pragma unroll
      for (int j = 0; j < 6; ++j) S11[j] += k1*R[j];
      const float kf = tw[352+32+c] * f2b*f2b;
      const float t1 = kf * d11 * (1.f/3.f);
      const float t2 = kf * d11 * d11 * (1.f/9.f);
      S11[0]+=t1*O[0]+t2; S11[1]+=t1*O[1]+t2; S11[2]+=t1*O[2]+t2;
      S11[3]+=t1*O[3];    S11[4]+=t1*O[4];    S11[5]+=t1*O[5];
      const float kc = tw[352+64+c] * f2c*f2c * (C222*C222);
      const float Qm[6] = {R[0]-trR3, R[1]-trR3, R[2]-trR3, R[3], R[4], R[5]};
      float R2[6]; symSq(Qm, R2);
#pragma unroll
      for (int j = 0; j < 6; ++j) S11[j] += kc*R2[j];
    }
    // (m2o,m2o,222)
    {
      const float kc = tw[448+c] * f2o*f2o * (C122*C122);
      float R2[6]; symSq(P, R2);
#pragma unroll
      for (int j = 0; j < 6; ++j) S12[j] += kc*R2[j];
    }
    // (m0e,m2e,022)
    {
      const float k1 = tw[480+c]    * m0a * f2a * u.u0;
      const float k2 = tw[480+32+c] * m0b * f2b;
      const float k3 = tw[480+64+c] * m0c * f2c * C222;
#pragma unroll
      for (int j = 0; j < 5; ++j) A13[j] += k1*u.s[j] + k2*qv[j] + k3*q5[j];
    }
  }

  // cross-half reductions + CG contractions + path weights
  const float s00 = RED(a0acc) * INV_SQRT96;
  const float s01 = RED(a1acc) * INV_SQRT3 * INV_SQRT64;
  const float s03 = RED(a3acc) * INV_SQRT5 * INV_SQRT96;
  const float s04 = RED(a4acc) * INV_SQRT5 * INV_SQRT32;
#pragma unroll
  for (int j = 0; j < 6; ++j) {
    S9[j] = RED(S9[j]); S11[j] = RED(S11[j]); S12[j] = RED(S12[j]);
  }
#pragma unroll
  for (int j = 0; j < 5; ++j) A13[j] = RED(A13[j]);

  float o9[5], o11[5], o12[5];
  vec5fromSym(S9,  o9);
  vec5fromSym(S11, o11);
  vec5fromSym(S12, o12);
  const float f0 = (s00*p0[0] + s01*p0[1] + s03*p0[3] + s04*p0[4]) * INV_SQRT5;
  float f2v[5];
#pragma unroll
  for (int j = 0; j < 5; ++j)
    f2v[j] = (o9[j]*INV_SQRT64*p2[0] + o11[j]*C222*INV_SQRT96*p2[2] +
              o12[j]*C222*INV_SQRT32*p2[3] + A13[j]*INV_SQRT96*p2[4]) *
             INV_SQRT5;

  // T = f0/sqrt(3)*I + mat3(f2)  (xyz), permuted to reference (y,z,x) order
  float Tm[6]; mat3from5(f2v, Tm);
  Tm[0] += f0*INV_SQRT3; Tm[1] += f0*INV_SQRT3; Tm[2] += f0*INV_SQRT3;
  const float T3[3][3] = {{Tm[0], Tm[3], Tm[5]},
                          {Tm[3], Tm[1], Tm[4]},
                          {Tm[5], Tm[4], Tm[2]}};
  if (half == 0 && (n0 + mrow) < N) {
    const int g = batch[n0 + mrow];
    const int perm[3] = {1, 2, 0};  // (y,z,x) -> xyz matrix index
#pragma unroll
    for (int a = 0; a < 3; ++a)
#pragma unroll
      for (int b = 0; b < 3; ++b)  // roll(+1,+1) applied at write time
        atomicAdd(&out[g*9 + ((a+1)%3)*3 + ((b+1)%3)], T3[perm[a]][perm[b]]);
  }
}

// ---------------------------------------------------------------------------
extern "C" void kernel_launch(void* const* d_in, const int* in_sizes, int n_in,
                              void* d_out, int out_size, void* d_ws,
                              size_t ws_size, hipStream_t stream) {
  (void)n_in; (void)d_ws; (void)ws_size;
  const float* node_scalar = (const float*)d_in[0];
  const float* node_equi   = (const float*)d_in[1];
  const int*   batch       = (const int*)d_in[2];
  const float* W0  = (const float*)d_in[3];
  const float* W1p = (const float*)d_in[4];
  const float* W2p = (const float*)d_in[5];
  const float* g0e = (const float*)d_in[6];
  const float* g1o = (const float*)d_in[7];
  // d_in[8] = g1e unused (m1e vanishes identically)
  const float* g2e = (const float*)d_in[9];
  const float* g2o = (const float*)d_in[10];
  const float* mw1 = (const float*)d_in[11];
  const float* mb1 = (const float*)d_in[12];
  const float* mw2 = (const float*)d_in[13];
  const float* mb2 = (const float*)d_in[14];
  const float* p0  = (const float*)d_in[15];
  // d_in[16] = p1 unused (f1 == 0 by antisymmetry)
  const float* p2  = (const float*)d_in[17];

  const int N = in_sizes[0] / 128;
  hipMemsetAsync(d_out, 0, (size_t)out_size * sizeof(float), stream);
  const int blocks = (N + 15) / 16;
  cart_tensor_kernel<<<blocks, TPB, 0, stream>>>(
      node_scalar, node_equi, batch, W0, W1p, W2p, g0e, g1o, g2e, g2o,
      mw1, mb1, mw2, mb2, p0, p2, (float*)d_out, N);
}